// Multi_Intere_Model_38835094291192
// MI455X (gfx1250) — compile-verified
//
#include <hip/hip_runtime.h>
#include <hip/hip_bf16.h>
#include <math.h>

// Problem constants (match reference)
constexpr int kB = 128;   // BATCH
constexpr int kL = 50;    // SEQ_L
constexpr int kT = 49;    // SEQ_L - 1
constexpr int kD = 64;    // EMBED_DIM
constexpr int kK = 4;     // INTERE_NUM
constexpr int kS = 256;   // SAMPLE_NUM

// LDS strides chosen for bank-conflict-free column access (stride % 64 == 4)
constexpr int ESTR = 68;
constexpr int GSTR = 68;
constexpr int WSTR = 52;  // K dim padded 50 -> 52 (13 wmma k-steps of 4)

typedef __attribute__((ext_vector_type(2))) float v2f;
typedef __attribute__((ext_vector_type(8))) float v8f;

// D = A(16x4) * B(4x16) + C, full f32 — precision-exact vs reference math.
__device__ __forceinline__ v8f wmma4(v2f a, v2f b, v8f c) {
  return __builtin_amdgcn_wmma_f32_16x16x4_f32(false, a, false, b, (short)0, c,
                                               false, false);
}

// ---------------------------------------------------------------------------
// Kernel 1: one block per batch row b.
//   - gather+normalize E_b (50x64) into LDS
//   - scores = E·Wq^T (50x4)
//   - G = E · targets^T via WMMA  (G[l][t] = E[l]·E[t+1])
//   - per 16-t chunk: softmax rows, sim[t][k] = Σ_l w·G, argmax k,
//     pos_score, and hitted = Whit × E via WMMA -> global workspace.
// ---------------------------------------------------------------------------
__global__ __launch_bounds__(256) void model_kernel(
    const int* __restrict__ seqs, const float* __restrict__ item_embed,
    const float* __restrict__ Wq, float* __restrict__ hitted_ws,
    float* __restrict__ pos_ws) {
  __shared__ float Elds[68][ESTR];   // rows 0..49 = normalized E, rest zero
  __shared__ float Glds[64][GSTR];   // G[l][t]
  __shared__ float Wch[64][WSTR];    // softmax weights for a 16-t chunk (rows = tc*4+k)
  __shared__ float Whit[16][WSTR];   // argmax-selected weight rows
  __shared__ float Slds[kL][kK];     // raw attention scores
  __shared__ float simL[16][kK];
  __shared__ int kmaxL[16];
  __shared__ float rnorm[kL];

  const int b = blockIdx.x;
  const int tid = threadIdx.x;
  const int lane = tid & 31;
  const int wave = tid >> 5;
  const int half = lane >> 4;   // selects K pair {0,1} vs {2,3} in a frag
  const int l16 = lane & 15;    // M (A-frag) / N (B,D-frag) index

  // --- zero-pad Elds, then gather raw rows ---
  for (int e = tid; e < 68 * ESTR; e += 256) (&Elds[0][0])[e] = 0.0f;
  __syncthreads();
  for (int e = tid; e < kL * kD; e += 256) {
    int l = e >> 6, d = e & 63;
    Elds[l][d] = item_embed[(size_t)seqs[b * kL + l] * kD + d];
  }
  __syncthreads();
  if (tid < kL) {
    float s = 0.f;
    for (int d = 0; d < kD; ++d) { float v = Elds[tid][d]; s += v * v; }
    rnorm[tid] = rsqrtf(s);
  }
  __syncthreads();
  for (int e = tid; e < kL * kD; e += 256) {
    int l = e >> 6, d = e & 63;
    Elds[l][d] *= rnorm[l];
  }
  __syncthreads();

  // --- attention scores (50x4), runs alongside G GEMM (both read-only on E) ---
  if (tid < kL * kK) {
    int l = tid >> 2, k = tid & 3;
    float s = 0.f;
    for (int d = 0; d < kD; ++d) s += Elds[l][d] * Wq[k * kD + d];
    Slds[l][k] = s;
  }

  // --- G GEMM: M=64 (l, padded), N=64 (t, padded), K=64; 16 tiles / 8 waves ---
  // A-frag layout (ISA 16x4 f32): lane%16 = M row; lanes<16 hold K0,K1; lanes>=16 K2,K3.
  // B-frag: lane%16 = N col; same K split.  B[k][n] = targets[n][k] = Elds[n+1][k].
  for (int tile = wave; tile < 16; tile += 8) {
    int mt = tile >> 2, nt = tile & 3;
    v8f acc = {};
    for (int kk = 0; kk < 16; ++kk) {
      int kb = kk * 4 + half * 2;
      v2f a, bb;
      a.x = Elds[mt * 16 + l16][kb];
      a.y = Elds[mt * 16 + l16][kb + 1];
      bb.x = Elds[nt * 16 + l16 + 1][kb];
      bb.y = Elds[nt * 16 + l16 + 1][kb + 1];
      acc = wmma4(a, bb, acc);
    }
    for (int i = 0; i < 8; ++i)   // D-frag: M = 8*half + i, N = lane%16
      Glds[mt * 16 + half * 8 + i][nt * 16 + l16] = acc[i];
  }
  __syncthreads();

  // --- process t in chunks of 16 ---
  for (int c = 0; c < 4; ++c) {
    // masked softmax rows: row r = tc*4+k  ->  w[t, l, k] over l<=t, zero-padded
    if (tid < 64) {
      int tc = tid >> 2, k = tid & 3;
      int t = c * 16 + tc;
      if (t < kT) {
        float mx = -3.4e38f;
        for (int l = 0; l <= t; ++l) mx = fmaxf(mx, Slds[l][k]);
        float sum = 0.f;
        for (int l = 0; l <= t; ++l) sum += expf(Slds[l][k] - mx);
        float inv = 1.0f / sum;
        int l = 0;
        for (; l <= t; ++l) Wch[tid][l] = expf(Slds[l][k] - mx) * inv;
        for (; l < WSTR; ++l) Wch[tid][l] = 0.f;
      } else {
        for (int l = 0; l < WSTR; ++l) Wch[tid][l] = 0.f;
      }
    }
    __syncthreads();
    // sim[t][k] = Σ_l w[t,l,k] * G[l][t]   (== intere_vec[t,k]·target[t])
    if (tid < 64) {
      int tc = tid >> 2, k = tid & 3;
      int t = c * 16 + tc;
      float s = 0.f;
      if (t < kT)
        for (int l = 0; l <= t; ++l) s += Wch[tid][l] * Glds[l][t];
      simL[tc][k] = s;
    }
    __syncthreads();
    // argmax over k; pos_score = sim at argmax (targets·hitted)
    if (tid < 16) {
      int t = c * 16 + tid;
      if (t < kT) {
        float best = simL[tid][0];
        int bk = 0;
        for (int k = 1; k < kK; ++k) {
          float v = simL[tid][k];
          if (v > best) { best = v; bk = k; }
        }
        kmaxL[tid] = bk;
        pos_ws[b * kT + t] = best;
      } else {
        kmaxL[tid] = 0;
      }
    }
    __syncthreads();
    for (int e = tid; e < 16 * WSTR; e += 256) {
      int tc = e / WSTR, l = e % WSTR;
      Whit[tc][l] = Wch[tc * 4 + kmaxL[tc]][l];  // invalid-t rows already zero
    }
    __syncthreads();
    // hitted GEMM: (16 x 52) x (52 x 64) — 4 N-tiles on waves 0..3, 13 k-steps
    if (wave < 4) {
      int nt = wave;
      v8f acc = {};
      for (int kk = 0; kk < 13; ++kk) {
        int kb = kk * 4 + half * 2;
        v2f a, bb;
        a.x = Whit[l16][kb];
        a.y = Whit[l16][kb + 1];
        bb.x = Elds[kb][nt * 16 + l16];
        bb.y = Elds[kb + 1][nt * 16 + l16];
        acc = wmma4(a, bb, acc);
      }
      for (int i = 0; i < 8; ++i) {
        int tc = half * 8 + i;
        int t = c * 16 + tc;
        if (t < kT)
          hitted_ws[((size_t)(b * kT + t)) * kD + nt * 16 + l16] = acc[i];
      }
    }
    __syncthreads();
  }
}

// ---------------------------------------------------------------------------
// Kernel 2: the bandwidth phase. One block per (b,t); one sample per thread.
// Recomputes the embedding norm on the fly (never materializes negE).
// ---------------------------------------------------------------------------
__global__ __launch_bounds__(256) void neg_kernel(
    const int* __restrict__ neg, const float* __restrict__ item_embed,
    const float* __restrict__ hitted_ws, const float* __restrict__ pos_ws,
    float* __restrict__ part_ws) {
  __shared__ __align__(16) float h[kD];
  __shared__ float red[256];
  const int bt = blockIdx.x;
  const int b = bt / kT;
  const int t = bt % kT;
  const int tid = threadIdx.x;
  if (tid < kD) h[tid] = hitted_ws[(size_t)bt * kD + tid];
  __syncthreads();

  const int idx = neg[((size_t)t * kB + b) * kS + tid];  // neg_samples[t][b][s]
  const float4* row = (const float4*)(item_embed + (size_t)idx * kD);
  const float4* h4 = (const float4*)h;
  float dot = 0.f, n2 = 0.f;
#pragma unroll
  for (int i = 0; i < kD / 4; ++i) {
    float4 v = row[i];
    float4 w = h4[i];
    dot += v.x * w.x + v.y * w.y + v.z * w.z + v.w * w.w;
    n2 += v.x * v.x + v.y * v.y + v.z * v.z + v.w * v.w;
  }
  float score = dot * rsqrtf(n2);

  // deterministic fixed-order logsumexp over 256 samples
  red[tid] = score;
  __syncthreads();
  for (int off = 128; off > 0; off >>= 1) {
    if (tid < off) red[tid] = fmaxf(red[tid], red[tid + off]);
    __syncthreads();
  }
  float mx = red[0];
  __syncthreads();
  red[tid] = expf(score - mx);
  __syncthreads();
  for (int off = 128; off > 0; off >>= 1) {
    if (tid < off) red[tid] += red[tid + off];
    __syncthreads();
  }
  if (tid == 0) part_ws[bt] = mx + logf(red[0]) - pos_ws[bt];
}

// ---------------------------------------------------------------------------
// Kernel 3: deterministic final reduction (no float atomics).
// ---------------------------------------------------------------------------
__global__ __launch_bounds__(256) void reduce_kernel(
    const float* __restrict__ part, float* __restrict__ out) {
  __shared__ float red[256];
  const int tid = threadIdx.x;
  float s = 0.f;
  for (int i = tid; i < kB * kT; i += 256) s += part[i];
  red[tid] = s;
  __syncthreads();
  for (int off = 128; off > 0; off >>= 1) {
    if (tid < off) red[tid] += red[tid + off];
    __syncthreads();
  }
  if (tid == 0) out[0] = red[0];
}

extern "C" void kernel_launch(void* const* d_in, const int* in_sizes, int n_in,
                              void* d_out, int out_size, void* d_ws,
                              size_t ws_size, hipStream_t stream) {
  const int* seqs = (const int*)d_in[0];
  const int* neg = (const int*)d_in[1];
  const float* item_embed = (const float*)d_in[2];
  const float* Wq = (const float*)d_in[3];
  float* out = (float*)d_out;

  // workspace carve-out (~1.66 MB)
  float* hitted_ws = (float*)d_ws;                       // kB*kT*kD
  float* pos_ws = hitted_ws + (size_t)kB * kT * kD;      // kB*kT
  float* part_ws = pos_ws + (size_t)kB * kT;             // kB*kT

  model_kernel<<<kB, 256, 0, stream>>>(seqs, item_embed, Wq, hitted_ws, pos_ws);
  neg_kernel<<<kB * kT, 256, 0, stream>>>(neg, item_embed, hitted_ws, pos_ws,
                                          part_ws);
  reduce_kernel<<<1, 256, 0, stream>>>(part_ws, out);
}